// SCGAT_33346126086584
// MI455X (gfx1250) — compile-verified
//
#include <hip/hip_runtime.h>
#include <math.h>

#define B_   32
#define C_   1024
#define T_   64
#define D_   64
#define F_   128
#define SDIM 64
#define NSTG 5
#define HMLP 64

typedef float v2f __attribute__((ext_vector_type(2)));
typedef float v8f __attribute__((ext_vector_type(8)));

// ---------------------------------------------------------------------------
// Kernel 0: L[i][j] = lam * log(A_init[i][j] + 1e-8)   (computed once, reused
// by all 32 batches -> avoids ~31M transcendentals)
// ---------------------------------------------------------------------------
__global__ void k_logA(const float* __restrict__ A, const float* __restrict__ lam,
                       float* __restrict__ Lm) {
    int idx = blockIdx.x * blockDim.x + threadIdx.x;
    if (idx < C_ * C_) Lm[idx] = lam[0] * logf(A[idx] + 1e-8f);
}

// ---------------------------------------------------------------------------
// Kernel 1: x_red[b,c,d] = mean_t x[b,c,t,d]   (streams the 512MB input,
// fully coalesced; this is the HBM roofline term ~22us at 23.3 TB/s)
// ---------------------------------------------------------------------------
__global__ void k_reduce_t(const float* __restrict__ x, float* __restrict__ xred) {
    int bc = blockIdx.x;                      // 0 .. B*C-1
    const float* xp = x + (size_t)bc * (T_ * D_);
    int tq = threadIdx.x >> 6;                // 0..3
    int d  = threadIdx.x & 63;
    float acc = 0.f;
    for (int t = tq; t < T_; t += 4) acc += xp[t * D_ + d];
    __shared__ float part[4][64];
    part[tq][d] = acc;
    __syncthreads();
    if (threadIdx.x < 64) {
        float s = part[0][threadIdx.x] + part[1][threadIdx.x] +
                  part[2][threadIdx.x] + part[3][threadIdx.x];
        xred[(size_t)bc * D_ + threadIdx.x] = s * (1.0f / T_);
    }
}

// ---------------------------------------------------------------------------
// Kernel 2: per-batch scalar MLP chain: x_glob -> s (softmax, output) ->
// c -> ce[b] = c . a_c.   256 threads: the C-reduction (the only sizable
// part) is split 4 ways to cut dependent-load latency; tail phases use 64.
// ---------------------------------------------------------------------------
__global__ void k_mlp(const float* __restrict__ xred,
                      const float* __restrict__ Ws1, const float* __restrict__ bs1,
                      const float* __restrict__ Ws2, const float* __restrict__ bs2,
                      const float* __restrict__ Wc1, const float* __restrict__ bc1,
                      const float* __restrict__ Wc2, const float* __restrict__ bc2,
                      const float* __restrict__ att,
                      float* __restrict__ s_out, float* __restrict__ ce) {
    int b = blockIdx.x;
    int tid = threadIdx.x;                    // 0..255
    __shared__ float part[4][64];
    __shared__ float xg[64], hid[64], sv[NSTG], ch[64], red[64];

    // x_glob = mean over C of x_red[b]  (4-way parallel over C)
    const float* xr = xred + (size_t)b * C_ * D_;
    int cq = tid >> 6, d = tid & 63;
    float acc = 0.f;
    for (int c = cq; c < C_; c += 4) acc += xr[c * D_ + d];
    part[cq][d] = acc;
    __syncthreads();
    if (tid < 64)
        xg[tid] = (part[0][tid] + part[1][tid] + part[2][tid] + part[3][tid]) *
                  (1.0f / C_);
    __syncthreads();

    // hid = relu(x_glob @ W_s1^T + b_s1)
    if (tid < 64) {
        float ha = bs1[tid];
        for (int dd = 0; dd < D_; ++dd) ha += xg[dd] * Ws1[tid * D_ + dd];
        hid[tid] = fmaxf(ha, 0.f);
    }
    __syncthreads();

    // logits -> softmax -> s
    if (tid < NSTG) {
        float l = bs2[tid];
        for (int h = 0; h < HMLP; ++h) l += hid[h] * Ws2[tid * HMLP + h];
        sv[tid] = l;
    }
    __syncthreads();
    if (tid == 0) {
        float mx = sv[0];
        for (int k = 1; k < NSTG; ++k) mx = fmaxf(mx, sv[k]);
        float sm = 0.f;
        for (int k = 0; k < NSTG; ++k) { sv[k] = __expf(sv[k] - mx); sm += sv[k]; }
        float inv = 1.0f / sm;
        for (int k = 0; k < NSTG; ++k) sv[k] *= inv;
    }
    __syncthreads();
    if (tid < NSTG) s_out[b * NSTG + tid] = sv[tid];

    // c = relu(s @ W_c1^T + b_c1) @ W_c2^T + b_c2 ; ce = c . a_c
    if (tid < 64) {
        float ca = bc1[tid];
        for (int k = 0; k < NSTG; ++k) ca += sv[k] * Wc1[tid * NSTG + k];
        ch[tid] = fmaxf(ca, 0.f);
    }
    __syncthreads();
    if (tid < 64) {
        float co = bc2[tid];
        for (int dd = 0; dd < SDIM; ++dd) co += ch[dd] * Wc2[tid * SDIM + dd];
        red[tid] = co * att[2 * F_ + tid];
    }
    __syncthreads();
    if (tid == 0) {
        float t = 0.f;
        for (int k = 0; k < 64; ++k) t += red[k];
        ce[b] = t;
    }
}

// ---------------------------------------------------------------------------
// Kernel 3: Z = x_red @ W_proj^T + b_proj  via f32 WMMA (16x16x4).
// One block per 16-row M tile; 8 waves = 8 N-tiles of F=128.
// ---------------------------------------------------------------------------
__global__ void k_zgemm(const float* __restrict__ xred, const float* __restrict__ Wp,
                        const float* __restrict__ bp, float* __restrict__ Z) {
    __shared__ float At[16 * 65];             // pad 65 -> bank-conflict free
    __shared__ float Wl[128 * 65];
    int m0  = blockIdx.x * 16;
    int tid = threadIdx.x;

    for (int idx = tid; idx < 16 * 64; idx += 256) {
        int r = idx >> 6, c = idx & 63;
        At[r * 65 + c] = xred[(size_t)(m0 + r) * 64 + c];
    }
    for (int idx = tid; idx < 128 * 64; idx += 256) {
        int f = idx >> 6, c = idx & 63;
        Wl[f * 65 + c] = Wp[idx];
    }
    __syncthreads();

    int wave = tid >> 5, lane = tid & 31;
    int n0 = wave * 16;
    int rr = lane & 15;
    int kk = (lane >> 4) * 2;                 // lanes 0-15: K0/K1, lanes 16-31: K2/K3
    int nn = rr;

    v8f acc = {};
    for (int k = 0; k < 64; k += 4) {
        v2f a, b;
        a.x = At[rr * 65 + k + kk];
        a.y = At[rr * 65 + k + kk + 1];
        b.x = Wl[(n0 + nn) * 65 + k + kk];
        b.y = Wl[(n0 + nn) * 65 + k + kk + 1];
        acc = __builtin_amdgcn_wmma_f32_16x16x4_f32(false, a, false, b,
                                                    (short)0, acc, false, false);
    }
    int g = lane >> 4;
    float bias = bp[n0 + nn];
    for (int v = 0; v < 8; ++v) {
        int row = v + 8 * g;                  // D layout: M = vgpr + 8*(lane>=16)
        Z[(size_t)(m0 + row) * F_ + n0 + nn] = acc[v] + bias;
    }
}

// ---------------------------------------------------------------------------
// Kernel 3b: ei = Z . a_i, ej = Z . a_j  (one wave per row, shuffle reduce)
// ---------------------------------------------------------------------------
__global__ void k_eij(const float* __restrict__ Z, const float* __restrict__ att,
                      float* __restrict__ ei, float* __restrict__ ej) {
    int wave = threadIdx.x >> 5, lane = threadIdx.x & 31;
    int m = blockIdx.x * 8 + wave;
    const float* zr = Z + (size_t)m * F_;
    float si = 0.f, sj = 0.f;
    for (int f = lane; f < F_; f += 32) {
        float z = zr[f];
        si += z * att[f];
        sj += z * att[F_ + f];
    }
    for (int o = 16; o >= 1; o >>= 1) {
        si += __shfl_xor(si, o, 32);
        sj += __shfl_xor(sj, o, 32);
    }
    if (lane == 0) { ei[m] = si; ej[m] = sj; }
}

// ---------------------------------------------------------------------------
// Kernel 4: fused attention: logits -> leakyrelu -> +L -> softmax -> h tile,
// alpha kept entirely in LDS (P stored transposed [j][row], exactly 64KB ->
// WMMA A-fragment reads are bank-conflict free). h tile via f32 WMMA, K=1024.
// Z stream prefetched 128 rows (32 iterations) ahead via global_prefetch_b8.
// ---------------------------------------------------------------------------
__global__ void k_attn(const float* __restrict__ Z, const float* __restrict__ ei,
                       const float* __restrict__ ej, const float* __restrict__ ce,
                       const float* __restrict__ Lm, float* __restrict__ h) {
    __shared__ float P[C_ * 16];              // P[j*16 + row], 64KB exactly
    int b  = blockIdx.x >> 6;
    int i0 = (blockIdx.x & 63) * 16;
    int row = threadIdx.x >> 4;               // 0..15
    int sub = threadIdx.x & 15;               // 0..15

    float eir = ei[b * C_ + i0 + row] + ce[b];
    const float* ejb = ej + (size_t)b * C_;
    const float* Lr  = Lm + (size_t)(i0 + row) * C_;

    // Pass 1: logits + running max (each thread touches only its own slots)
    float m = -3.4e38f;
    for (int j = sub; j < C_; j += 16) {
        float e = eir + ejb[j];
        e = (e > 0.f) ? e : 0.2f * e;         // leaky_relu(0.2)
        e += Lr[j];
        P[j * 16 + row] = e;
        m = fmaxf(m, e);
    }
    for (int o = 8; o >= 1; o >>= 1) m = fmaxf(m, __shfl_xor(m, o, 16));

    // Pass 2: exp + sum; Pass 3: fold 1/sum into P (so WMMA output is final)
    float s = 0.f;
    for (int j = sub; j < C_; j += 16) {
        float p = __expf(P[j * 16 + row] - m);
        P[j * 16 + row] = p;
        s += p;
    }
    for (int o = 8; o >= 1; o >>= 1) s += __shfl_xor(s, o, 16);
    float inv = 1.0f / s;
    for (int j = sub; j < C_; j += 16) P[j * 16 + row] *= inv;
    __syncthreads();

    // WMMA phase: 8 waves, each a 16x16 f-tile; K = 1024 in steps of 4
    int wave = threadIdx.x >> 5, lane = threadIdx.x & 31;
    int n0 = wave * 16;
    int rr = lane & 15;
    int kk = (lane >> 4) * 2;
    int nn = rr;
    const float* Zb = Z + (size_t)b * C_ * F_;

    v8f acc = {};
    for (int k = 0; k < C_; k += 4) {
        if (k + 130 < C_) {                   // uniform guard: EXEC stays all-ones
            // pull next L2 lines toward the WGP 32 iterations ahead
            __builtin_prefetch(&Zb[(size_t)(k + kk + 128) * F_ + n0 + nn], 0, 0);
        }
        v2f a, bb;
        a.x  = P[(k + kk) * 16 + rr];         // lanes = consecutive rows: no bank conflicts
        a.y  = P[(k + kk + 1) * 16 + rr];
        bb.x = Zb[(size_t)(k + kk) * F_ + n0 + nn];
        bb.y = Zb[(size_t)(k + kk + 1) * F_ + n0 + nn];
        acc = __builtin_amdgcn_wmma_f32_16x16x4_f32(false, a, false, bb,
                                                    (short)0, acc, false, false);
    }
    int g = lane >> 4;
    for (int v = 0; v < 8; ++v) {
        int r2 = v + 8 * g;
        h[((size_t)b * C_ + i0 + r2) * F_ + n0 + nn] = acc[v];
    }
}

// ---------------------------------------------------------------------------
extern "C" void kernel_launch(void* const* d_in, const int* in_sizes, int n_in,
                              void* d_out, int out_size, void* d_ws, size_t ws_size,
                              hipStream_t stream) {
    const float* x    = (const float*)d_in[0];
    const float* A    = (const float*)d_in[1];
    const float* Wp   = (const float*)d_in[2];
    const float* bp   = (const float*)d_in[3];
    const float* Ws1  = (const float*)d_in[4];
    const float* bs1  = (const float*)d_in[5];
    const float* Ws2  = (const float*)d_in[6];
    const float* bs2  = (const float*)d_in[7];
    const float* Wc1  = (const float*)d_in[8];
    const float* bc1  = (const float*)d_in[9];
    const float* Wc2  = (const float*)d_in[10];
    const float* bc2  = (const float*)d_in[11];
    const float* att  = (const float*)d_in[12];
    const float* lam  = (const float*)d_in[13];

    float* h_out = (float*)d_out;                          // (B,C,F)
    float* s_out = h_out + (size_t)B_ * C_ * F_;           // (B,NSTG)

    // workspace layout (~30 MB)
    float* xred = (float*)d_ws;                            // B*C*D
    float* Z    = xred + (size_t)B_ * C_ * D_;             // B*C*F
    float* Lm   = Z    + (size_t)B_ * C_ * F_;             // C*C
    float* ei   = Lm   + (size_t)C_ * C_;                  // B*C
    float* ej   = ei   + (size_t)B_ * C_;                  // B*C
    float* ce   = ej   + (size_t)B_ * C_;                  // B

    k_logA    <<<(C_ * C_ + 255) / 256, 256, 0, stream>>>(A, lam, Lm);
    k_reduce_t<<<B_ * C_,              256, 0, stream>>>(x, xred);
    k_mlp     <<<B_,                   256, 0, stream>>>(xred, Ws1, bs1, Ws2, bs2,
                                                         Wc1, bc1, Wc2, bc2, att,
                                                         s_out, ce);
    k_zgemm   <<<(B_ * C_) / 16,       256, 0, stream>>>(xred, Wp, bp, Z);
    k_eij     <<<(B_ * C_) / 8,        256, 0, stream>>>(Z, att, ei, ej);
    k_attn    <<<B_ * (C_ / 16),       256, 0, stream>>>(Z, ei, ej, ce, Lm, h_out);
}